// InductiveModel_16862041604204
// MI455X (gfx1250) — compile-verified
//
#include <hip/hip_runtime.h>
#include <hip/hip_bf16.h>

typedef __attribute__((ext_vector_type(2))) float v2f;
typedef __attribute__((ext_vector_type(8))) float v8f;

#define DIN 128  // all three SAGE layers have in-dim 128

// ---------------------------------------------------------------- utilities
__global__ void zero_f32_kernel(float* __restrict__ p, long n) {
    long i = (long)blockIdx.x * blockDim.x + threadIdx.x;
    long stride = (long)gridDim.x * blockDim.x;
    for (; i < n; i += stride) p[i] = 0.0f;
}

__global__ void degree_count_kernel(const long long* __restrict__ dst,
                                    float* __restrict__ cnt, int E) {
    int e = blockIdx.x * blockDim.x + threadIdx.x;
    if (e < E) atomicAdd(&cnt[(int)dst[e]], 1.0f);
}

__global__ void inv_degree_kernel(float* __restrict__ cnt, int N) {
    int i = blockIdx.x * blockDim.x + threadIdx.x;
    if (i < N) cnt[i] = 1.0f / fmaxf(cnt[i], 1.0f);
}

// ------------------------------------------------------- scatter-add (sum)
// One thread handles a float4 chunk of one edge's message (128 -> 32 chunks).
// Atomic traffic resolves in L2 (agg is 25.6 MB, L2 is 192 MB).
__global__ void scatter_add_kernel(const float* __restrict__ h,
                                   const long long* __restrict__ src,
                                   const long long* __restrict__ dst,
                                   float* __restrict__ agg, int E) {
    const int chunks = DIN >> 2;  // 32
    int t = blockIdx.x * blockDim.x + threadIdx.x;
    int e = t / chunks;
    if (e >= E) return;
    int c = (t - e * chunks) << 2;
    int s = (int)src[e];
    int d = (int)dst[e];
    const float4 v = *(const float4*)(h + (long)s * DIN + c);
    float* a = agg + (long)d * DIN + c;
    atomicAdd(a + 0, v.x);
    atomicAdd(a + 1, v.y);
    atomicAdd(a + 2, v.z);
    atomicAdd(a + 3, v.w);
}

// ------------------------------------- fused SAGE GEMM epilogue via WMMA f32
// out[M, dout] = (agg[m,:] * inv[m]) @ Wl + bl + x @ Wr   (+ optional ReLU)
//
// Block = 256 threads (8 waves). Each block owns one 16-column tile and
// stages the [DIN x 16] slabs of Wl and Wr into LDS once (16 KB total);
// its 8 waves then grid-stride over 16-row tiles, reading B fragments from
// LDS (conflict-free [k][n] layout) and A fragments as contiguous float2
// global loads. One V_WMMA_F32_16X16X4_F32 per matrix per K-step.
__global__ __launch_bounds__(256)
void sage_gemm_wmma_kernel(const float* __restrict__ agg,   // [N, DIN]
                           const float* __restrict__ inv,   // [N]
                           const float* __restrict__ x,     // [N, DIN]
                           const float* __restrict__ Wl,    // [DIN, dout]
                           const float* __restrict__ bl,    // [dout]
                           const float* __restrict__ Wr,    // [DIN, dout]
                           float* __restrict__ out,         // [N, dout]
                           int dout, int relu, int mtiles) {
    __shared__ float sWl[DIN * 16];   // [k][n] slab of Wl for this column tile
    __shared__ float sWr[DIN * 16];   // [k][n] slab of Wr

    const int col0 = blockIdx.y * 16;

    // ---- stage weight slabs to LDS (coalesced over n within each k-row)
    for (int idx = threadIdx.x; idx < DIN * 16; idx += 256) {
        const int k = idx >> 4;
        const int n = idx & 15;
        const long g = (long)k * dout + col0 + n;
        sWl[idx] = Wl[g];
        sWr[idx] = Wr[g];
    }
    __syncthreads();

    const int lane = threadIdx.x & 31;
    const int wave = threadIdx.x >> 5;
    const int ln   = lane & 15;              // fragment row/col within tile
    const int koff = (lane >> 4) << 1;       // 0 for lanes 0-15, 2 for 16-31
    const float bias = bl[col0 + ln];

    for (int mt = blockIdx.x * 8 + wave; mt < mtiles; mt += gridDim.x * 8) {
        const int row0 = mt * 16;
        const int m    = row0 + ln;
        const float scale = inv[m];          // mean normalization, folded in
        const float* __restrict__ arow = agg + (long)m * DIN;
        const float* __restrict__ xrow = x   + (long)m * DIN;

        v8f c = {};
#pragma unroll 8
        for (int k0 = 0; k0 < DIN; k0 += 4) {
            const int k = k0 + koff;
            v2f a, b, xa, xb;
            // A fragment (16x4): lane<16 -> K=k0..k0+1, lane>=16 -> K=k0+2..3
            a.x  = arow[k]     * scale;
            a.y  = arow[k + 1] * scale;
            xa.x = xrow[k];
            xa.y = xrow[k + 1];
            // B fragment (4x16) from LDS, [k][n] layout -> conflict-free
            b.x  = sWl[k * 16 + ln];
            b.y  = sWl[(k + 1) * 16 + ln];
            xb.x = sWr[k * 16 + ln];
            xb.y = sWr[(k + 1) * 16 + ln];
            c = __builtin_amdgcn_wmma_f32_16x16x4_f32(false, a,  false, b,
                                                      (short)0, c, false, false);
            c = __builtin_amdgcn_wmma_f32_16x16x4_f32(false, xa, false, xb,
                                                      (short)0, c, false, false);
        }

        const int mbase = row0 + ((lane >> 4) << 3);  // +8 rows for lanes 16-31
#pragma unroll
        for (int r = 0; r < 8; ++r) {
            float v = c[r] + bias;
            if (relu) v = fmaxf(v, 0.0f);
            out[(long)(mbase + r) * dout + col0 + ln] = v;
        }
    }
}

// ---------------------------------------------------------------- launcher
extern "C" void kernel_launch(void* const* d_in, const int* in_sizes, int n_in,
                              void* d_out, int out_size, void* d_ws, size_t ws_size,
                              hipStream_t stream) {
    const float*     x   = (const float*)d_in[0];
    const long long* ei  = (const long long*)d_in[1];
    const float* Wl0 = (const float*)d_in[2];
    const float* bl0 = (const float*)d_in[3];
    const float* Wr0 = (const float*)d_in[4];
    const float* Wl1 = (const float*)d_in[5];
    const float* bl1 = (const float*)d_in[6];
    const float* Wr1 = (const float*)d_in[7];
    const float* Wl2 = (const float*)d_in[8];
    const float* bl2 = (const float*)d_in[9];
    const float* Wr2 = (const float*)d_in[10];

    const int N = in_sizes[0] / DIN;   // 50000
    const int E = in_sizes[1] / 2;     // 800000
    const long long* src = ei;
    const long long* dst = ei + E;

    float* inv = (float*)d_ws;                  // [N]
    float* agg = inv + N;                       // [N, 128]
    float* hA  = agg + (size_t)N * DIN;         // [N, 128]
    float* hB  = hA  + (size_t)N * DIN;         // [N, 128]
    float* out = (float*)d_out;                 // [N, 64]

    const int ZB = 1024;  // blocks for grid-stride zero fill
    const long nAgg = (long)N * DIN;
    const int mtiles = N / 16;                  // 3125 (N divisible by 16)

    // ---- degrees -> reciprocal mean scale (shared across layers)
    zero_f32_kernel<<<ZB, 256, 0, stream>>>(inv, (long)N);
    degree_count_kernel<<<(E + 255) / 256, 256, 0, stream>>>(dst, inv, E);
    inv_degree_kernel<<<(N + 255) / 256, 256, 0, stream>>>(inv, N);

    const int scatterBlocks = (E * (DIN / 4) + 255) / 256;
    const dim3 gemmBlock(256, 1, 1);
    const int gemmGX = 98;  // 98 blocks * 8 waves ~ 784 waves per column tile

    // ---- layer 0: 128 -> 128, ReLU
    zero_f32_kernel<<<ZB, 256, 0, stream>>>(agg, nAgg);
    scatter_add_kernel<<<scatterBlocks, 256, 0, stream>>>(x, src, dst, agg, E);
    sage_gemm_wmma_kernel<<<dim3(gemmGX, 128 / 16), gemmBlock, 0, stream>>>(
        agg, inv, x, Wl0, bl0, Wr0, hA, 128, 1, mtiles);

    // ---- layer 1: 128 -> 128, ReLU
    zero_f32_kernel<<<ZB, 256, 0, stream>>>(agg, nAgg);
    scatter_add_kernel<<<scatterBlocks, 256, 0, stream>>>(hA, src, dst, agg, E);
    sage_gemm_wmma_kernel<<<dim3(gemmGX, 128 / 16), gemmBlock, 0, stream>>>(
        agg, inv, hA, Wl1, bl1, Wr1, hB, 128, 1, mtiles);

    // ---- layer 2: 128 -> 64, no activation
    zero_f32_kernel<<<ZB, 256, 0, stream>>>(agg, nAgg);
    scatter_add_kernel<<<scatterBlocks, 256, 0, stream>>>(hB, src, dst, agg, E);
    sage_gemm_wmma_kernel<<<dim3(gemmGX, 64 / 16), gemmBlock, 0, stream>>>(
        agg, inv, hB, Wl2, bl2, Wr2, out, 64, 0, mtiles);
}